// DistanceAutoMLCriterion_26250840113742
// MI455X (gfx1250) — compile-verified
//
#include <hip/hip_runtime.h>
#include <hip/hip_bf16.h>
#include <math.h>

typedef __attribute__((ext_vector_type(2))) float v2f;
typedef __attribute__((ext_vector_type(4))) float v4f;
typedef __attribute__((ext_vector_type(8))) float v8f;

#define EPS 1e-8f
#define PAD 0

// ---------------------------------------------------------------------------
// Phase 1: per-row max / argmax / nll over V=32000 logits. Pure streaming.
// One block per row, 256 threads, float4 non-temporal loads (1 GB total).
// ---------------------------------------------------------------------------
__global__ __launch_bounds__(256)
void rowmax_kernel(const float* __restrict__ pred_ll,
                   const int* __restrict__ target,
                   float* __restrict__ wmax,
                   int* __restrict__ warg,
                   float* __restrict__ wnll,
                   int V) {
    __shared__ float sm[256];
    __shared__ int si[256];

    const int row = blockIdx.x;
    const int tid = threadIdx.x;
    const float* rp = pred_ll + (size_t)row * (size_t)V;

    float m = -__builtin_inff();
    int idx = -1;

    const int nvec = V >> 2;
    const v4f* rp4 = (const v4f*)rp;
    // Each thread visits strictly increasing columns -> strict '>' keeps the
    // first occurrence within a thread; cross-thread ties resolved in reduce.
    for (int c = tid; c < nvec; c += 256) {
        v4f v = __builtin_nontemporal_load(rp4 + c);
        const int base = c << 2;
        if (v.x > m) { m = v.x; idx = base; }
        if (v.y > m) { m = v.y; idx = base + 1; }
        if (v.z > m) { m = v.z; idx = base + 2; }
        if (v.w > m) { m = v.w; idx = base + 3; }
    }
    for (int c = (nvec << 2) + tid; c < V; c += 256) {   // generic tail
        float v = rp[c];
        if (v > m) { m = v; idx = c; }
    }

    sm[tid] = m;
    si[tid] = idx;
    __syncthreads();
    for (int s = 128; s > 0; s >>= 1) {
        if (tid < s) {
            const float m2 = sm[tid + s];
            const int i2 = si[tid + s];
            // tie -> smallest index (argmax first-occurrence semantics);
            // unsigned compare makes idx==-1 lose.
            if (m2 > sm[tid] || (m2 == sm[tid] && (unsigned)i2 < (unsigned)si[tid])) {
                sm[tid] = m2;
                si[tid] = i2;
            }
        }
        __syncthreads();
    }
    if (tid == 0) {
        wmax[row] = sm[0];
        warg[row] = si[0];
        const int t = target[row];
        wnll[row] = -rp[t];
    }
}

// ---------------------------------------------------------------------------
// Phase 2: cosine(gold_e, pred_e) + loss combine, 16 rows per wave via
// V_WMMA_F32_16X16X4_F32. Diagonals of (G x P^T), (G x G^T), (P x P^T)
// give dot/norm^2 per row at full fp32 precision.
//
// f32 16x4 A-layout: lane l holds row m=l&15, k = 2*(l>>4) + reg.
// f32 4x16 B-layout is the same mapping of the transposed matrix, so one
// float2 load per lane per matrix serves either operand slot.
// ---------------------------------------------------------------------------
__global__ __launch_bounds__(32)
void cosine_loss_kernel(const float* __restrict__ weight,
                        const int* __restrict__ target,
                        const float* __restrict__ alpha,
                        const float* __restrict__ beta,
                        const float* __restrict__ wmax,
                        const int* __restrict__ warg,
                        const float* __restrict__ wnll,
                        float* __restrict__ partial,   // [gridDim.x * 2]
                        int D, int N) {
    __shared__ float tile0[256];
    __shared__ float tile1[256];
    __shared__ float tile2[256];

    const int lane = threadIdx.x;          // 0..31
    const int r16 = blockIdx.x * 16;
    const int mrow = lane & 15;            // row within the 16-row tile
    const int koff = (lane >> 4) * 2;      // k sub-offset per A/B layout
    int grow = r16 + mrow;
    if (grow >= N) grow = N - 1;           // clamp (masked out below)

    const int tgt = target[grow];
    const int pos = warg[grow];
    const float* gp = weight + (size_t)tgt * (size_t)D;
    const float* pp = weight + (size_t)pos * (size_t)D;

    v8f acc_gp = {};   // diag -> dot(gold, pred)
    v8f acc_gg = {};   // diag -> ||gold||^2
    v8f acc_pp = {};   // diag -> ||pred||^2

    for (int k = 0; k < D; k += 4) {
        const v2f g = *(const v2f*)(gp + k + koff);
        const v2f p = *(const v2f*)(pp + k + koff);
        acc_gp = __builtin_amdgcn_wmma_f32_16x16x4_f32(
            false, g, false, p, (short)0, acc_gp, false, false);
        acc_gg = __builtin_amdgcn_wmma_f32_16x16x4_f32(
            false, g, false, g, (short)0, acc_gg, false, false);
        acc_pp = __builtin_amdgcn_wmma_f32_16x16x4_f32(
            false, p, false, p, (short)0, acc_pp, false, false);
    }

    // C/D layout: VGPR j holds M = j + 8*(lane>=16), N = lane&15.
    const int Mbase = (lane >> 4) * 8;
    const int Ncol = lane & 15;
#pragma unroll
    for (int j = 0; j < 8; ++j) {
        const int off = (Mbase + j) * 16 + Ncol;
        tile0[off] = acc_gp[j];
        tile1[off] = acc_gg[j];
        tile2[off] = acc_pp[j];
    }
    __syncthreads();

    float s_loss = 0.0f, s_nll = 0.0f;
    if (lane < 16 && (r16 + lane) < N) {
        const int d = lane * 17;           // diagonal (i,i)
        const float dot = tile0[d];
        const float gg = tile1[d];
        const float qq = tile2[d];
        const float na = fmaxf(sqrtf(gg), EPS);
        const float nb = fmaxf(sqrtf(qq), EPS);
        const float distance = dot / (na * nb);
        const float a = alpha[tgt];
        const float b = beta[tgt];
        const float x = 0.5f / (1.0f + expf(-(a * distance + b)));  // sigmoid/2
        const float nll = wnll[r16 + lane];
        const float pred_loss = -wmax[r16 + lane];
        const float maskv = (tgt != PAD) ? 1.0f : 0.0f;
        const float loss = (x + 0.5f) * nll + (0.5f - x) * pred_loss;
        s_loss = loss * maskv;
        s_nll = nll * maskv;
    }
    // wave32 reduce (lanes 16..31 contribute zero)
    for (int off = 16; off > 0; off >>= 1) {
        s_loss += __shfl_down(s_loss, off, 32);
        s_nll  += __shfl_down(s_nll,  off, 32);
    }
    if (lane == 0) {
        partial[blockIdx.x * 2 + 0] = s_loss;
        partial[blockIdx.x * 2 + 1] = s_nll;
    }
}

// ---------------------------------------------------------------------------
// Phase 3: deterministic fixed-order reduction of partials into d_out[0..1].
// ---------------------------------------------------------------------------
__global__ __launch_bounds__(256)
void final_reduce_kernel(const float* __restrict__ partial, int nb,
                         float* __restrict__ out) {
    __shared__ float s0[256];
    __shared__ float s1[256];
    const int tid = threadIdx.x;
    float a = 0.0f, b = 0.0f;
    for (int i = tid; i < nb; i += 256) {
        a += partial[2 * i + 0];
        b += partial[2 * i + 1];
    }
    s0[tid] = a;
    s1[tid] = b;
    __syncthreads();
    for (int s = 128; s > 0; s >>= 1) {
        if (tid < s) { s0[tid] += s0[tid + s]; s1[tid] += s1[tid + s]; }
        __syncthreads();
    }
    if (tid == 0) {
        out[0] = s0[0];
        out[1] = s1[0];
    }
}

extern "C" void kernel_launch(void* const* d_in, const int* in_sizes, int n_in,
                              void* d_out, int out_size, void* d_ws, size_t ws_size,
                              hipStream_t stream) {
    const float* pred_ll = (const float*)d_in[0];
    const int*   target  = (const int*)d_in[1];
    const float* weight  = (const float*)d_in[2];
    const float* alpha   = (const float*)d_in[3];
    const float* beta    = (const float*)d_in[4];

    const int N = in_sizes[1];              // 8192 rows
    const int V = in_sizes[3];              // 32000 vocab
    const int D = in_sizes[2] / V;          // 512 embedding dim
    float* out = (float*)d_out;

    char* ws = (char*)d_ws;
    float* wmax    = (float*)(ws);
    int*   warg    = (int*)  (ws + (size_t)N * 4);
    float* wnll    = (float*)(ws + (size_t)N * 8);
    float* partial = (float*)(ws + (size_t)N * 12);

    const int nb2 = (N + 15) / 16;

    rowmax_kernel<<<N, 256, 0, stream>>>(pred_ll, target, wmax, warg, wnll, V);
    cosine_loss_kernel<<<nb2, 32, 0, stream>>>(weight, target, alpha, beta,
                                               wmax, warg, wnll, partial, D, N);
    final_reduce_kernel<<<1, 256, 0, stream>>>(partial, nb2, out);
}